// PointTDA_31610959298727
// MI455X (gfx1250) — compile-verified
//
#include <hip/hip_runtime.h>
#include <math.h>

// Problem constants (match reference)
#define BB 16
#define NN 8192
#define SS 1024
#define KK 32
#define EPS_BN 1e-5f
#define FACTOR 1.0f

typedef __attribute__((ext_vector_type(2))) float v2f;
typedef __attribute__((ext_vector_type(8))) float v8f;

// ---------------------------------------------------------------------------
// 0) zero the small stats accumulators in workspace
// ---------------------------------------------------------------------------
__global__ void init_stats_kernel(float* stats) {
    if (threadIdx.x < 16) stats[threadIdx.x] = 0.0f;
}

// ---------------------------------------------------------------------------
// 1) Farthest point sampling: one workgroup per batch, distance[] in LDS,
//    xyz stays L2-resident. 1024 sequential rounds of min-update + argmax.
//    Writes new_xyz[b][s][3] (the carry-in farthest, idx[0]=0, as in ref).
// ---------------------------------------------------------------------------
__global__ void fps_kernel(const float* __restrict__ xyz, float* __restrict__ new_xyz) {
    __shared__ float dist[NN];          // 32 KB of the 320 KB WGP LDS
    __shared__ float redV[32];
    __shared__ int   redI[32];
    __shared__ int   farIdx;

    const float* p  = xyz + (size_t)blockIdx.x * NN * 3;
    float*       nx = new_xyz + (size_t)blockIdx.x * SS * 3;

    for (int i = threadIdx.x; i < NN; i += blockDim.x) dist[i] = 1e10f;
    if (threadIdx.x == 0) farIdx = 0;
    __syncthreads();

    const int wave = threadIdx.x >> 5;
    const int lane = threadIdx.x & 31;

    for (int s = 0; s < SS; ++s) {
        const int far = farIdx;
        const float cx = p[far * 3 + 0];
        const float cy = p[far * 3 + 1];
        const float cz = p[far * 3 + 2];
        if (threadIdx.x == 0) {
            nx[s * 3 + 0] = cx; nx[s * 3 + 1] = cy; nx[s * 3 + 2] = cz;
        }
        __syncthreads();  // everyone has read farIdx before it is overwritten

        float best = -1.0f; int bidx = 0x7fffffff;
        for (int i = threadIdx.x; i < NN; i += blockDim.x) {
            const float dx = p[i * 3 + 0] - cx;
            const float dy = p[i * 3 + 1] - cy;
            const float dz = p[i * 3 + 2] - cz;
            const float d  = dx * dx + dy * dy + dz * dz;
            const float dm = fminf(dist[i], d);
            dist[i] = dm;
            if (dm > best || (dm == best && i < bidx)) { best = dm; bidx = i; }
        }
        // wave32 butterfly argmax (first-index tie-break, like jnp.argmax)
        #pragma unroll
        for (int off = 16; off > 0; off >>= 1) {
            const float ov = __shfl_xor(best, off, 32);
            const int   oi = __shfl_xor(bidx, off, 32);
            if (ov > best || (ov == best && oi < bidx)) { best = ov; bidx = oi; }
        }
        if (lane == 0) { redV[wave] = best; redI[wave] = bidx; }
        __syncthreads();
        if (threadIdx.x < 32) {
            best = redV[threadIdx.x]; bidx = redI[threadIdx.x];
            #pragma unroll
            for (int off = 16; off > 0; off >>= 1) {
                const float ov = __shfl_xor(best, off, 32);
                const int   oi = __shfl_xor(bidx, off, 32);
                if (ov > best || (ov == best && oi < bidx)) { best = ov; bidx = oi; }
            }
            if (threadIdx.x == 0) farIdx = bidx;
        }
        __syncthreads();
    }
}

// ---------------------------------------------------------------------------
// 2) Distance tiles via V_WMMA_F32_16X16X4_F32, one batch at a time so the
//    32 MB sq tile stays resident in the 192 MB L2 for the selection pass.
//    sq'[s][n] = |p_n|^2 - 2 q_s . p_n   (row-constant |q|^2 dropped: it does
//    not affect per-row top-K ordering).
//    A row m = (-2qx,-2qy,-2qz, 1) ; B col n = (px,py,pz,|p|^2).
//    grid (4, 64), block 256: wave = one 16-query x 256-point strip.
// ---------------------------------------------------------------------------
__global__ void dist_wmma_kernel(const float* __restrict__ xyz_b,
                                 const float* __restrict__ newxyz_b,
                                 float* __restrict__ sq) {
    const int wave = threadIdx.x >> 5;
    const int lane = threadIdx.x & 31;
    const int sTile = blockIdx.y;              // 0..63
    const int strip = blockIdx.x * 8 + wave;   // 0..31, each strip = 16 N-tiles

    // A operand: 16x4 f32, lanes 0-15 hold K={0,1}, lanes 16-31 hold K={2,3}
    const int m = lane & 15;
    const int srow = sTile * 16 + m;
    const float qx = newxyz_b[srow * 3 + 0];
    const float qy = newxyz_b[srow * 3 + 1];
    const float qz = newxyz_b[srow * 3 + 2];
    v2f a;
    if (lane < 16) { a.x = -2.0f * qx; a.y = -2.0f * qy; }
    else           { a.x = -2.0f * qz; a.y = 1.0f; }

    for (int t = 0; t < 16; ++t) {
        const int n0 = (strip * 16 + t) * 16;
        const int n  = n0 + (lane & 15);
        const float px = xyz_b[n * 3 + 0];
        const float py = xyz_b[n * 3 + 1];
        const float pz = xyz_b[n * 3 + 2];
        const float pn = px * px + py * py + pz * pz;
        // B operand: 4x16 f32, VGPR0 = rows {K=0 | K=2}, VGPR1 = rows {K=1 | K=3}
        v2f bm;
        if (lane < 16) { bm.x = px; bm.y = py; }
        else           { bm.x = pz; bm.y = pn; }

        v8f c = {};
        c = __builtin_amdgcn_wmma_f32_16x16x4_f32(
                /*neg_a=*/false, a, /*neg_b=*/false, bm,
                /*c_mod=*/(short)0, c, /*reuse_a=*/false, /*reuse_b=*/false);

        // D layout: VGPR r -> lanes 0-15: M=r, lanes 16-31: M=r+8; N = lane%16
        const int col = n0 + (lane & 15);
        const int rowBase = sTile * 16 + ((lane < 16) ? 0 : 8);
        #pragma unroll
        for (int r = 0; r < 8; ++r) {
            sq[(size_t)(rowBase + r) * NN + col] = c[r];
        }
    }
}

// ---------------------------------------------------------------------------
// 3) Top-K=32 selection per query: one wave per query row. 32 rounds of
//    wave-wide argmin over the L2-resident row; winner is poisoned with +inf
//    (store fenced with s_wait_storecnt so the same wave's re-reads see it).
// ---------------------------------------------------------------------------
__global__ void knn_select_kernel(float* __restrict__ sq, int* __restrict__ knn_b) {
    const int wave = threadIdx.x >> 5;
    const int lane = threadIdx.x & 31;
    const int q = blockIdx.x * 8 + wave;       // 0..1023
    float* row = sq + (size_t)q * NN;

    for (int k = 0; k < KK; ++k) {
        float best = 3.0e38f; int bidx = 0x7fffffff;
        for (int i = lane; i < NN; i += 32) {
            const float v = row[i];
            if (v < best || (v == best && i < bidx)) { best = v; bidx = i; }
        }
        #pragma unroll
        for (int off = 16; off > 0; off >>= 1) {
            const float ov = __shfl_xor(best, off, 32);
            const int   oi = __shfl_xor(bidx, off, 32);
            if (ov < best || (ov == best && oi < bidx)) { best = ov; bidx = oi; }
        }
        if (lane == 0) {
            knn_b[q * KK + k] = bidx;
            row[bidx] = 3.0e38f;               // exclude for later rounds
        }
        // make the poison store visible to this wave's next-round loads
        asm volatile("s_wait_storecnt 0x0" ::: "memory");
    }
}

// ---------------------------------------------------------------------------
// 4) Gather grouped points, form diff = p[knn] - q, accumulate global
//    sum/sumsq for the scalar unbiased std. One thread per (b,s,k).
// ---------------------------------------------------------------------------
__global__ void group_kernel(const float* __restrict__ xyz,
                             const float* __restrict__ new_xyz,
                             const int* __restrict__ knn,
                             float* __restrict__ diffb,
                             float* __restrict__ stats) {
    __shared__ float rs[8], rq[8];
    const int tid = blockIdx.x * blockDim.x + threadIdx.x;   // == b*S*K + s*K + k
    const int b = tid / (SS * KK);
    const int s = (tid / KK) % SS;
    const int idx = knn[tid];

    const float dx = xyz[((size_t)b * NN + idx) * 3 + 0] - new_xyz[((size_t)b * SS + s) * 3 + 0];
    const float dy = xyz[((size_t)b * NN + idx) * 3 + 1] - new_xyz[((size_t)b * SS + s) * 3 + 1];
    const float dz = xyz[((size_t)b * NN + idx) * 3 + 2] - new_xyz[((size_t)b * SS + s) * 3 + 2];
    diffb[(size_t)tid * 3 + 0] = dx;
    diffb[(size_t)tid * 3 + 1] = dy;
    diffb[(size_t)tid * 3 + 2] = dz;

    float lsum = dx + dy + dz;
    float lsq  = dx * dx + dy * dy + dz * dz;
    #pragma unroll
    for (int off = 16; off > 0; off >>= 1) {
        lsum += __shfl_xor(lsum, off, 32);
        lsq  += __shfl_xor(lsq,  off, 32);
    }
    const int wave = threadIdx.x >> 5, lane = threadIdx.x & 31;
    if (lane == 0) { rs[wave] = lsum; rq[wave] = lsq; }
    __syncthreads();
    if (threadIdx.x == 0) {
        float ts = 0.f, tq = 0.f;
        for (int w = 0; w < 8; ++w) { ts += rs[w]; tq += rq[w]; }
        atomicAdd(&stats[0], ts);
        atomicAdd(&stats[1], tq);
    }
}

// ---------------------------------------------------------------------------
// 5) Pool over K: one wave per (b,s). lc[c] = max_k + mean_k of channel c,
//    channels = [diff/ (std+1e-5) (3), center (3)]; center => 2*center.
//    Also accumulates per-channel BN1 statistics.
// ---------------------------------------------------------------------------
__global__ void pool_kernel(const float* __restrict__ diffb,
                            const float* __restrict__ new_xyz,
                            float* __restrict__ lc,
                            float* __restrict__ stats) {
    const int wave = threadIdx.x >> 5;
    const int lane = threadIdx.x & 31;     // lane == k
    const int gw = blockIdx.x * 8 + wave;  // 0 .. B*S-1
    const int b = gw / SS;
    const int s = gw % SS;

    const float nTot = (float)(BB * SS * KK * 3);
    const float sum = stats[0], sumsq = stats[1];
    const float var = (sumsq - sum * sum / nTot) / (nTot - 1.0f);
    const float denom = sqrtf(var) + 1e-5f;

    const size_t dbase = ((size_t)gw * KK + lane) * 3;
    float v[6];
    #pragma unroll
    for (int c = 0; c < 3; ++c) {
        const float x = diffb[dbase + c] / denom;
        float mx = x, sm = x;
        #pragma unroll
        for (int off = 16; off > 0; off >>= 1) {
            mx = fmaxf(mx, __shfl_xor(mx, off, 32));
            sm += __shfl_xor(sm, off, 32);
        }
        v[c] = mx + sm * (1.0f / KK);
    }
    #pragma unroll
    for (int c = 0; c < 3; ++c) {
        const float ctr = new_xyz[((size_t)b * SS + s) * 3 + c];
        v[3 + c] = 2.0f * ctr;             // max + mean of a constant
    }
    if (lane == 0) {
        #pragma unroll
        for (int c = 0; c < 6; ++c) {
            lc[((size_t)b * 6 + c) * SS + s] = v[c];
            atomicAdd(&stats[2 + c], v[c]);
            atomicAdd(&stats[8 + c], v[c] * v[c]);
        }
    }
}

// ---------------------------------------------------------------------------
// 6) BN1+relu (recomputed on the fly) and zero-safe cdist over S=1024:
//    one workgroup per (b, channel-pair).
// ---------------------------------------------------------------------------
__global__ void cdist_kernel(const float* __restrict__ lc,
                             const float* __restrict__ stats,
                             const float* __restrict__ g1,
                             const float* __restrict__ b1,
                             float* __restrict__ feat) {
    __shared__ float red[256];
    const int i = blockIdx.x / 6;
    const int j = blockIdx.x % 6;
    const int b = blockIdx.y;

    const float n = (float)(BB * SS);
    const float mi = stats[2 + i] / n;
    const float vi = stats[8 + i] / n - mi * mi;
    const float mj = stats[2 + j] / n;
    const float vj = stats[8 + j] / n - mj * mj;
    const float ri = rsqrtf(vi + EPS_BN);
    const float rj = rsqrtf(vj + EPS_BN);
    const float gi = g1[i], bi = b1[i], gj = g1[j], bj = b1[j];

    float acc = 0.0f;
    for (int s = threadIdx.x; s < SS; s += 256) {
        const float xi = lc[((size_t)b * 6 + i) * SS + s];
        const float xj = lc[((size_t)b * 6 + j) * SS + s];
        const float yi = fmaxf((xi - mi) * ri * gi + bi, 0.0f);
        const float yj = fmaxf((xj - mj) * rj * gj + bj, 0.0f);
        const float d = yi - yj;
        acc += d * d;
    }
    red[threadIdx.x] = acc;
    __syncthreads();
    for (int off = 128; off > 0; off >>= 1) {
        if (threadIdx.x < off) red[threadIdx.x] += red[threadIdx.x + off];
        __syncthreads();
    }
    if (threadIdx.x == 0) {
        const float sqv = red[0];
        float val = (sqv > 0.0f) ? sqrtf(sqv) : 0.0f;
        if (j == i + 1) val *= FACTOR;
        feat[b * 36 + i * 6 + j] = val;
    }
}

// ---------------------------------------------------------------------------
// 7) BN2 over the batch dimension + relu -> d_out [B,36]
// ---------------------------------------------------------------------------
__global__ void bn2_kernel(const float* __restrict__ feat,
                           const float* __restrict__ g2,
                           const float* __restrict__ b2,
                           float* __restrict__ out) {
    const int f = threadIdx.x;
    if (f >= 36) return;
    float m = 0.0f;
    for (int b = 0; b < BB; ++b) m += feat[b * 36 + f];
    m *= (1.0f / BB);
    float v = 0.0f;
    for (int b = 0; b < BB; ++b) {
        const float d = feat[b * 36 + f] - m;
        v += d * d;
    }
    v *= (1.0f / BB);
    const float inv = rsqrtf(v + EPS_BN);
    for (int b = 0; b < BB; ++b) {
        out[b * 36 + f] = fmaxf((feat[b * 36 + f] - m) * inv * g2[f] + b2[f], 0.0f);
    }
}

// ---------------------------------------------------------------------------
// Host orchestration
// ---------------------------------------------------------------------------
extern "C" void kernel_launch(void* const* d_in, const int* in_sizes, int n_in,
                              void* d_out, int out_size, void* d_ws, size_t ws_size,
                              hipStream_t stream) {
    const float* xyz = (const float*)d_in[0];   // [B,N,3]
    const float* g1  = (const float*)d_in[1];   // [6]
    const float* b1  = (const float*)d_in[2];   // [6]
    const float* g2  = (const float*)d_in[3];   // [36]
    const float* b2  = (const float*)d_in[4];   // [36]
    float* out = (float*)d_out;                 // [B,36]

    // workspace layout (floats); ~41 MB total
    float* ws = (float*)d_ws;
    float* new_xyz = ws;                                    // B*S*3     = 49152
    float* sq      = new_xyz + (size_t)BB * SS * 3;         // S*N       = 8388608 (one batch, reused)
    int*   knn     = (int*)(sq + (size_t)SS * NN);          // B*S*K     = 524288 ints
    float* diffb   = (float*)(knn + (size_t)BB * SS * KK);  // B*S*K*3   = 1572864
    float* lc      = diffb + (size_t)BB * SS * KK * 3;      // B*6*S     = 98304
    float* stats   = lc + (size_t)BB * 6 * SS;              // 16
    float* feat    = stats + 16;                            // B*36      = 576

    init_stats_kernel<<<1, 64, 0, stream>>>(stats);
    fps_kernel<<<BB, 1024, 0, stream>>>(xyz, new_xyz);

    // per-batch: WMMA distance tile (32 MB, L2-resident) then top-K select
    for (int b = 0; b < BB; ++b) {
        dist_wmma_kernel<<<dim3(4, 64), 256, 0, stream>>>(
            xyz + (size_t)b * NN * 3, new_xyz + (size_t)b * SS * 3, sq);
        knn_select_kernel<<<128, 256, 0, stream>>>(sq, knn + (size_t)b * SS * KK);
    }

    group_kernel<<<(BB * SS * KK) / 256, 256, 0, stream>>>(xyz, new_xyz, knn, diffb, stats);
    pool_kernel<<<(BB * SS) / 8, 256, 0, stream>>>(diffb, new_xyz, lc, stats);
    cdist_kernel<<<dim3(36, BB), 256, 0, stream>>>(lc, stats, g1, b1, feat);
    bn2_kernel<<<1, 64, 0, stream>>>(feat, g2, b2, out);
}